// Model_86706799772343
// MI455X (gfx1250) — compile-verified
//
#include <hip/hip_runtime.h>

typedef float v2f __attribute__((ext_vector_type(2)));
typedef float v8f __attribute__((ext_vector_type(8)));

#define LSTM_T 4096
#define LSTM_B 1024
#define XSTRIDE (LSTM_B * 4)   // elements per timestep row
#define L2E 1.44269504088896340736f
#define PF 8                   // prefetch block (steps); 2*PF steps per outer iter

// Exchange with lane^16 (wave32): ds_swizzle group-of-32, xor=0x10, and=0x1f
__device__ __forceinline__ float swap16(float x) {
  int i = __builtin_amdgcn_ds_swizzle(__builtin_bit_cast(int, x), 0x401F);
  return __builtin_bit_cast(float, i);
}

// sigmoid(x) = rcp(1 + exp2(-x*log2e))  (no trailing fma)
__device__ __forceinline__ float sigm(float v) {
  return __builtin_amdgcn_rcpf(1.0f + __builtin_amdgcn_exp2f(v * (-L2E)));
}

// fma(rcp(1+exp2(k1*v)), k2, k3): tanh -> k1=-2log2e, k2=2, k3=-1
__device__ __forceinline__ float actf(float v, float k1, float k2, float k3) {
  float e = __builtin_amdgcn_exp2f(v * k1);
  float r = __builtin_amdgcn_rcpf(1.0f + e);
  return __builtin_fmaf(r, k2, k3);
}

__device__ __forceinline__ float tanh_fast(float v) {
  return actf(v, -2.0f * L2E, 2.0f, -1.0f);
}

// Gate-row permutation: new order [i0..3, g0..3, f0..3, o0..3] over
// PyTorch order [i, f, g, o]. Group map 0,1,2,3 -> 0,2,1,3.
__device__ __forceinline__ int perm_row(int m) {
  int grp = m >> 2;
  int og  = ((grp & 1) << 1) | (grp >> 1);
  return og * 4 + (m & 3);
}

struct LstmState {
  v2f bh;                    // B(h) operand: VGPR0={h0 lo,h2 hi}, VGPR1={h1 lo,h3 hi}
  float c0, c1, c2, c3;      // cell state, duplicated in both half-waves
};

__device__ __forceinline__ void lstm_step(
    v2f bx, v2f a_ih, v2f a_hh, v8f cbias, int hi,
    float g1, float g2, float g3, LstmState& st, float* po) {
  // gates = W_ih * x_t^T + bias ; += W_hh * h_{t-1}^T
  v8f d = __builtin_amdgcn_wmma_f32_16x16x4_f32(false, a_ih, false, bx,
                                                (short)0, cbias, false, false);
  d = __builtin_amdgcn_wmma_f32_16x16x4_f32(false, a_hh, false, st.bh,
                                            (short)0, d, false, false);

  // VGPRs 0-3: i (lo) / f (hi) -> sigmoid.  VGPRs 4-7: g (lo, tanh) / o (hi, sigmoid)
  float v0 = sigm(d[0]);
  float v1 = sigm(d[1]);
  float v2 = sigm(d[2]);
  float v3 = sigm(d[3]);
  float v4 = actf(d[4], g1, g2, g3);
  float v5 = actf(d[5], g1, g2, g3);
  float v6 = actf(d[6], g1, g2, g3);
  float v7 = actf(d[7], g1, g2, g3);

  // Early cross-lane: low half needs o0, o1 (high's v4, v5); overlaps q/c work
  float s4 = swap16(v4);
  float s5 = swap16(v5);

  // lo half: q = i*g ; hi half: q = f*c ; c_new = q + q^(lane^16), both halves
  float q0 = v0 * (hi ? st.c0 : v4);
  float q1 = v1 * (hi ? st.c1 : v5);
  float q2 = v2 * (hi ? st.c2 : v6);
  float q3 = v3 * (hi ? st.c3 : v7);
  st.c0 = q0 + swap16(q0);
  st.c1 = q1 + swap16(q1);
  st.c2 = q2 + swap16(q2);
  st.c3 = q3 + swap16(q3);

  // Each half only ever consumes two h values:
  //   lo: h0 = o0*tanh(c0), h1 = o1*tanh(c1)   (o0,o1 via s4,s5)
  //   hi: h2 = o2*tanh(c2), h3 = o3*tanh(c3)   (o2,o3 = local v6,v7)
  float ca = hi ? st.c2 : st.c0;
  float cb = hi ? st.c3 : st.c1;
  float oa = hi ? v6 : s4;
  float ob = hi ? v7 : s5;
  st.bh[0] = oa * tanh_fast(ca);
  st.bh[1] = ob * tanh_fast(cb);

  // Branch-free store: lo lane -> {h0,h1} at po, hi lane -> {h2,h3} at po+2
  // (po already includes the +2*hi offset); covers the 256B tile row coalesced.
  __builtin_nontemporal_store(st.bh, (v2f*)po);
}

__global__ __launch_bounds__(32) void lstm_wmma_kernel(
    const float* __restrict__ x,    // [T, B, 4]
    const float* __restrict__ Wih,  // [16, 4]
    const float* __restrict__ Whh,  // [16, 4]
    const float* __restrict__ bih,  // [16]
    const float* __restrict__ bhh,  // [16]
    float* __restrict__ out) {      // [T, B, 4]
  const int lane = threadIdx.x;      // 0..31, one wave per block
  const int hi   = lane >> 4;
  const int col  = lane & 15;
  const int b    = (blockIdx.x << 4) + col;
  const int k0   = hi << 1;          // K pair held by this half (A layout)

  // A operands (16x4 f32): lane m holds W[perm(m)][k0..k0+1]
  const int mo = perm_row(col);
  v2f a_ih = *(const v2f*)(Wih + mo * 4 + k0);
  v2f a_hh = *(const v2f*)(Whh + mo * 4 + k0);

  // C bias: VGPR j = gate row j + 8*hi, constant across columns
  v8f cbias;
#pragma unroll
  for (int j = 0; j < 8; ++j) {
    int r = perm_row(j + (hi << 3));
    cbias[j] = bih[r] + bhh[r];
  }

  // Per-lane activation constants for VGPRs 4-7: tanh (lo) / sigmoid (hi)
  const float g1 = hi ? -L2E : -2.0f * L2E;
  const float g2 = hi ? 1.0f : 2.0f;
  const float g3 = hi ? 0.0f : -1.0f;

  LstmState st;
  st.bh = (v2f){0.0f, 0.0f};
  st.c0 = st.c1 = st.c2 = st.c3 = 0.0f;

  const int xoff = b * 4 + k0;       // per-lane element offset in a timestep row
  const float* pb = x + xoff;        // block base for t=0

  // Preload block A (t = 0..PF-1)
  v2f bufA[PF], bufB[PF];
#pragma unroll
  for (int j = 0; j < PF; ++j)
    bufA[j] = __builtin_nontemporal_load((const v2f*)(pb + j * XSTRIDE));
  pb += PF * XSTRIDE;                // pb -> t = to + PF at loop top

  // Store pointer with the per-half +2 offset folded in once
  float* po = out + (size_t)(b * 4 + (hi << 1));

  for (int to = 0; to < LSTM_T; to += 2 * PF) {
    // Prefetch block B (t = to+PF .. to+2PF-1): always in range
#pragma unroll
    for (int j = 0; j < PF; ++j)
      bufB[j] = __builtin_nontemporal_load((const v2f*)(pb + j * XSTRIDE));

    // Process block A while block B is in flight
#pragma unroll
    for (int j = 0; j < PF; ++j) {
      lstm_step(bufA[j], a_ih, a_hh, cbias, hi, g1, g2, g3, st, po);
      po += XSTRIDE;
    }

    // Prefetch next block A (t = to+2PF ..): clamp the final OOB block to t=0
    const float* pa = (to + 3 * PF <= LSTM_T) ? (pb + PF * XSTRIDE) : (x + xoff);
#pragma unroll
    for (int j = 0; j < PF; ++j)
      bufA[j] = __builtin_nontemporal_load((const v2f*)(pa + j * XSTRIDE));

    // Process block B
#pragma unroll
    for (int j = 0; j < PF; ++j) {
      lstm_step(bufB[j], a_ih, a_hh, cbias, hi, g1, g2, g3, st, po);
      po += XSTRIDE;
    }

    pb += 2 * PF * XSTRIDE;
  }
}

extern "C" void kernel_launch(void* const* d_in, const int* in_sizes, int n_in,
                              void* d_out, int out_size, void* d_ws, size_t ws_size,
                              hipStream_t stream) {
  const float* input = (const float*)d_in[0];
  const float* W_ih  = (const float*)d_in[1];
  const float* W_hh  = (const float*)d_in[2];
  const float* b_ih  = (const float*)d_in[3];
  const float* b_hh  = (const float*)d_in[4];
  float* out = (float*)d_out;
  (void)in_sizes; (void)n_in; (void)d_ws; (void)ws_size; (void)out_size;

  // One wave (32 threads) per 16-batch tile: B/16 = 64 blocks.
  lstm_wmma_kernel<<<dim3(LSTM_B / 16), dim3(32), 0, stream>>>(
      input, W_ih, W_hh, b_ih, b_hh, out);
}